// GAT_90692529422659
// MI455X (gfx1250) — compile-verified
//
#include <hip/hip_runtime.h>
#include <hip/hip_bf16.h>
#include <math.h>

// ---------------- problem constants (from reference) ----------------
#define NNODES 50000
#define FIN    256
#define H1     8
#define C1     8
#define HC1    64     // H1*C1
#define NC     40

typedef __attribute__((ext_vector_type(2))) float v2f;
typedef __attribute__((ext_vector_type(8))) float v8f;

// ---------------- float atomic max via sign-split trick ----------------
__device__ __forceinline__ void atomicMaxFloat(float* addr, float value) {
    if (value >= 0.0f) {
        atomicMax((int*)addr, __float_as_int(value));
    } else {
        atomicMin((unsigned int*)addr, __float_as_uint(value));
    }
}

// ---------------- WMMA fp32 GEMM: D[M x NCOL] = A[M x K] * B[K x NCOL] ----------------
// One wave (32 threads) per 16-row M tile; NT = ceil(NCOL/16) accumulators.
// Uses V_WMMA_F32_16X16X4_F32; K must be a multiple of 4, M a multiple of 16.
template<int K, int NCOL>
__global__ __launch_bounds__(32)
void wmma_gemm_f32(const float* __restrict__ A,
                   const float* __restrict__ B,
                   float* __restrict__ D) {
    constexpr int NT = (NCOL + 15) / 16;
    const int tile_m = blockIdx.x;
    const int lane   = threadIdx.x;      // 0..31
    const int half   = lane >> 4;        // 0 or 1
    const int l16    = lane & 15;

    const float* __restrict__ arow = A + (size_t)(tile_m * 16 + l16) * K;

    v8f acc[NT] = {};

    #pragma unroll 2
    for (int k0 = 0; k0 < K; k0 += 4) {
        const int ka = k0 + 2 * half;            // this lane's K pair
        v2f a;
        a.x = arow[ka];
        a.y = arow[ka + 1];
        #pragma unroll
        for (int t = 0; t < NT; ++t) {
            const int col = t * 16 + l16;
            v2f b;
            if (col < NCOL) {
                b.x = B[(size_t)ka * NCOL + col];
                b.y = B[(size_t)(ka + 1) * NCOL + col];
            } else {
                b.x = 0.0f; b.y = 0.0f;
            }
            acc[t] = __builtin_amdgcn_wmma_f32_16x16x4_f32(
                false, a, false, b, (short)0, acc[t], false, false);
        }
    }

    // C/D layout: VGPR r -> row = r + 8*half, col = l16 (per 16x16 tile)
    #pragma unroll
    for (int t = 0; t < NT; ++t) {
        const int col = t * 16 + l16;
        if (col < NCOL) {
            #pragma unroll
            for (int r = 0; r < 8; ++r) {
                const int orow = tile_m * 16 + r + 8 * half;
                D[(size_t)orow * NCOL + col] = acc[t][r];
            }
        }
    }
}

// ---------------- per-node: attention logits + init of segment buffers ----------------
template<int HEADS, int CH>
__global__ void node_alpha_init(const float* __restrict__ hfeat,   // [N, HEADS*CH]
                                const float* __restrict__ a_src,   // [HEADS*CH]
                                const float* __restrict__ a_dst,   // [HEADS*CH]
                                float* __restrict__ as,            // [N, HEADS]
                                float* __restrict__ ad,            // [N, HEADS]
                                float* __restrict__ m,             // [N, HEADS] -> -inf
                                float* __restrict__ s,             // [N, HEADS] -> 0
                                float* __restrict__ acc,           // [N, HEADS*CH] -> 0
                                int n_nodes) {
    int n = blockIdx.x * blockDim.x + threadIdx.x;
    if (n >= n_nodes) return;
    const float* hp = hfeat + (size_t)n * HEADS * CH;
    #pragma unroll
    for (int h = 0; h < HEADS; ++h) {
        float sa = 0.0f, sd = 0.0f;
        #pragma unroll
        for (int c = 0; c < CH; ++c) {
            float v = hp[h * CH + c];
            sa += v * a_src[h * CH + c];
            sd += v * a_dst[h * CH + c];
        }
        as[(size_t)n * HEADS + h] = sa;
        ad[(size_t)n * HEADS + h] = sd;
        m [(size_t)n * HEADS + h] = -INFINITY;
        s [(size_t)n * HEADS + h] = 0.0f;
    }
    float* ap = acc + (size_t)n * HEADS * CH;
    #pragma unroll
    for (int i = 0; i < HEADS * CH; ++i) ap[i] = 0.0f;
}

// ---------------- edge pass 1: segment max of leaky-relu logits ----------------
template<int HEADS>
__global__ void edge_max_k(const int* __restrict__ src, const int* __restrict__ dst,
                           int e_tot, int e_orig,
                           const float* __restrict__ as, const float* __restrict__ ad,
                           float* __restrict__ m) {
    int e = blockIdx.x * blockDim.x + threadIdx.x;
    if (e >= e_tot) return;
    int si, di;
    if (e < e_orig) { si = src[e]; di = dst[e]; }
    else            { si = di = e - e_orig; }          // self loops
    #pragma unroll
    for (int h = 0; h < HEADS; ++h) {
        float t = as[(size_t)si * HEADS + h] + ad[(size_t)di * HEADS + h];
        t = (t > 0.0f) ? t : 0.2f * t;                 // leaky_relu(0.2)
        atomicMaxFloat(&m[(size_t)di * HEADS + h], t);
    }
}

// ---------------- edge pass 2: p=exp(e-m); s += p; acc += p * h[src] ----------------
template<int HEADS, int CH>
__global__ void edge_accum_k(const int* __restrict__ src, const int* __restrict__ dst,
                             int e_tot, int e_orig,
                             const float* __restrict__ as, const float* __restrict__ ad,
                             const float* __restrict__ m,
                             const float* __restrict__ hfeat,
                             float* __restrict__ s, float* __restrict__ acc) {
    int e = blockIdx.x * blockDim.x + threadIdx.x;
    if (e >= e_tot) return;
    int si, di;
    if (e < e_orig) { si = src[e]; di = dst[e]; }
    else            { si = di = e - e_orig; }
    float p[HEADS];
    #pragma unroll
    for (int h = 0; h < HEADS; ++h) {
        float t = as[(size_t)si * HEADS + h] + ad[(size_t)di * HEADS + h];
        t = (t > 0.0f) ? t : 0.2f * t;
        p[h] = __expf(t - m[(size_t)di * HEADS + h]);
        atomicAdd(&s[(size_t)di * HEADS + h], p[h]);
    }
    const float* hp = hfeat + (size_t)si * HEADS * CH;
    float* ap = acc + (size_t)di * HEADS * CH;
    #pragma unroll
    for (int h = 0; h < HEADS; ++h) {
        #pragma unroll
        for (int c = 0; c < CH; ++c) {
            atomicAdd(&ap[h * CH + c], p[h] * hp[h * CH + c]);
        }
    }
}

// ---------------- layer-1 finish: normalize, +bias, ELU ----------------
__global__ void finish1_k(const float* __restrict__ acc, const float* __restrict__ s,
                          const float* __restrict__ bias, float* __restrict__ out,
                          int n_nodes) {
    int n = blockIdx.x * blockDim.x + threadIdx.x;
    if (n >= n_nodes) return;
    const float* ap = acc + (size_t)n * HC1;
    float* op = out + (size_t)n * HC1;
    #pragma unroll
    for (int h = 0; h < H1; ++h) {
        float denom = s[(size_t)n * H1 + h] + 1e-16f;
        #pragma unroll
        for (int c = 0; c < C1; ++c) {
            float v = ap[h * C1 + c] / denom + bias[h * C1 + c];
            op[h * C1 + c] = (v > 0.0f) ? v : (__expf(v) - 1.0f);   // ELU
        }
    }
}

// ---------------- layer-2 finish: normalize, +bias, log_softmax -> d_out ----------------
__global__ void finish2_k(const float* __restrict__ acc, const float* __restrict__ s,
                          const float* __restrict__ bias, float* __restrict__ out,
                          int n_nodes) {
    int n = blockIdx.x * blockDim.x + threadIdx.x;
    if (n >= n_nodes) return;
    float denom = s[n] + 1e-16f;
    float logit[NC];
    float mx = -INFINITY;
    #pragma unroll
    for (int c = 0; c < NC; ++c) {
        logit[c] = acc[(size_t)n * NC + c] / denom + bias[c];
        mx = fmaxf(mx, logit[c]);
    }
    float sum = 0.0f;
    #pragma unroll
    for (int c = 0; c < NC; ++c) sum += __expf(logit[c] - mx);
    float lse = mx + logf(sum);
    #pragma unroll
    for (int c = 0; c < NC; ++c) out[(size_t)n * NC + c] = logit[c] - lse;
}

// ---------------- host launch ----------------
extern "C" void kernel_launch(void* const* d_in, const int* in_sizes, int n_in,
                              void* d_out, int out_size, void* d_ws, size_t ws_size,
                              hipStream_t stream) {
    const float* x      = (const float*)d_in[0];
    const int*   esrc   = (const int*)  d_in[1];
    const int*   edst   = (const int*)  d_in[2];
    const float* W1     = (const float*)d_in[3];
    const float* a1s    = (const float*)d_in[4];
    const float* a1d    = (const float*)d_in[5];
    const float* b1     = (const float*)d_in[6];
    const float* W2     = (const float*)d_in[7];
    const float* a2s    = (const float*)d_in[8];
    const float* a2d    = (const float*)d_in[9];
    const float* b2     = (const float*)d_in[10];

    const int n_nodes = in_sizes[0] / FIN;     // 50000
    const int n_edges = in_sizes[1];           // 800000
    const int e_tot   = n_edges + n_nodes;     // with self loops

    // workspace carve-up (floats)
    float* ws = (float*)d_ws;
    float* h1    = ws;                 ws += (size_t)n_nodes * HC1;
    float* as1   = ws;                 ws += (size_t)n_nodes * H1;
    float* ad1   = ws;                 ws += (size_t)n_nodes * H1;
    float* m1    = ws;                 ws += (size_t)n_nodes * H1;
    float* s1    = ws;                 ws += (size_t)n_nodes * H1;
    float* acc1  = ws;                 ws += (size_t)n_nodes * HC1;
    float* act1  = ws;                 ws += (size_t)n_nodes * HC1;
    float* h2    = ws;                 ws += (size_t)n_nodes * NC;
    float* as2   = ws;                 ws += (size_t)n_nodes;
    float* ad2   = ws;                 ws += (size_t)n_nodes;
    float* m2    = ws;                 ws += (size_t)n_nodes;
    float* s2    = ws;                 ws += (size_t)n_nodes;
    float* acc2  = ws;                 ws += (size_t)n_nodes * NC;

    float* out = (float*)d_out;

    const int NB  = (n_nodes + 255) / 256;
    const int EB  = (e_tot + 255) / 256;
    const int MT  = n_nodes / 16;      // 3125 M tiles, exact

    // ---- layer 1 ----
    wmma_gemm_f32<FIN, HC1><<<MT, 32, 0, stream>>>(x, W1, h1);
    node_alpha_init<H1, C1><<<NB, 256, 0, stream>>>(h1, a1s, a1d, as1, ad1, m1, s1, acc1, n_nodes);
    edge_max_k<H1><<<EB, 256, 0, stream>>>(esrc, edst, e_tot, n_edges, as1, ad1, m1);
    edge_accum_k<H1, C1><<<EB, 256, 0, stream>>>(esrc, edst, e_tot, n_edges, as1, ad1, m1, h1, s1, acc1);
    finish1_k<<<NB, 256, 0, stream>>>(acc1, s1, b1, act1, n_nodes);

    // ---- layer 2 ----
    wmma_gemm_f32<HC1, NC><<<MT, 32, 0, stream>>>(act1, W2, h2);
    node_alpha_init<1, NC><<<NB, 256, 0, stream>>>(h2, a2s, a2d, as2, ad2, m2, s2, acc2, n_nodes);
    edge_max_k<1><<<EB, 256, 0, stream>>>(esrc, edst, e_tot, n_edges, as2, ad2, m2);
    edge_accum_k<1, NC><<<EB, 256, 0, stream>>>(esrc, edst, e_tot, n_edges, as2, ad2, m2, h2, s2, acc2);
    finish2_k<<<NB, 256, 0, stream>>>(acc2, s2, b2, out, n_nodes);
}